// ZINC_GIN_Outer_9534827397804
// MI455X (gfx1250) — compile-verified
//
#include <hip/hip_runtime.h>
#include <hip/hip_bf16.h>

// ---------------------------------------------------------------------------
// ZINC GIN outer model on MI455X (gfx1250). All GEMMs run on the matrix pipe
// via V_WMMA_F32_16X16X4_F32 (full fp32 precision; workload is memory/atomic
// bound so the f32 WMMA rate suffices once issue overhead is amortized).
// Weights are pre-transposed so every B fragment is a single contiguous
// global_load_b64; heavy kernels use M=32 blocking per wave so each B
// fragment feeds two WMMAs. Bond-encoder + message + scatter are fused so
// the 328MB E x H tensor is never materialized.
// ---------------------------------------------------------------------------

typedef float v2f __attribute__((ext_vector_type(2)));
typedef float v8f __attribute__((ext_vector_type(8)));

#define NN 40000
#define EE 640000
#define GG 1024
#define HH 128
#define FIN 28

__device__ __forceinline__ v8f wmma4(v2f a, v2f b, v8f c) {
  // D = A(16x4 f32) x B(4x16 f32) + C(16x16 f32)
  return __builtin_amdgcn_wmma_f32_16x16x4_f32(false, a, false, b, (short)0, c,
                                               false, false);
}

__device__ __forceinline__ v8f v8zero() {
  v8f z = {0.f, 0.f, 0.f, 0.f, 0.f, 0.f, 0.f, 0.f};
  return z;
}

// ------------------------------ utility ------------------------------------

__global__ void k_zero(float* __restrict__ p, int n) {
  int i = blockIdx.x * blockDim.x + threadIdx.x;
  if (i < n) p[i] = 0.f;
}

// dst[c*K + k] = src[k*C + c]   (K x C row-major -> C x K "B-fragment" layout)
__global__ void k_transpose(const float* __restrict__ src,
                            float* __restrict__ dst, int K, int C) {
  int idx = blockIdx.x * blockDim.x + threadIdx.x;
  if (idx < K * C) {
    int k = idx / C, c = idx % C;
    dst[c * K + k] = src[idx];
  }
}

// -------------------------- encoder weight fusion ---------------------------
// W1ft[c*28+i] = (aw @ mw_top)[i][c] ; W2ft[c*128+i] = (ew @ mw_bot)[i][c]
// bf[c]        = ab @ mw_top + eb @ mw_bot + mb     (outputs pre-transposed)
__global__ void k_fuse_weights(const float* __restrict__ aw,
                               const float* __restrict__ ab,
                               const float* __restrict__ ew,
                               const float* __restrict__ eb,
                               const float* __restrict__ mw,
                               const float* __restrict__ mb,
                               float* __restrict__ W1ft,
                               float* __restrict__ W2ft,
                               float* __restrict__ bf) {
  int tid = blockIdx.x * blockDim.x + threadIdx.x;
  if (tid < FIN * HH) {
    int i = tid / HH, c = tid % HH;
    float s = 0.f;
    for (int j = 0; j < 64; ++j) s += aw[i * 64 + j] * mw[j * HH + c];
    W1ft[c * FIN + i] = s;
  } else if (tid < FIN * HH + HH * HH) {
    int t = tid - FIN * HH;
    int i = t / HH, c = t % HH;
    float s = 0.f;
    for (int j = 0; j < 64; ++j) s += ew[i * 64 + j] * mw[(64 + j) * HH + c];
    W2ft[c * HH + i] = s;
  } else if (tid < FIN * HH + HH * HH + HH) {
    int c = tid - FIN * HH - HH * HH;
    float s = mb[c];
    for (int j = 0; j < 64; ++j) s += ab[j] * mw[j * HH + c];
    for (int j = 0; j < 64; ++j) s += eb[j] * mw[(64 + j) * HH + c];
    bf[c] = s;
  }
}

// ------------------------------ encoder -------------------------------------
// h0 = relu(x @ W1f + inter @ W2f + bf)   one 16-node tile per wave
__global__ void __launch_bounds__(128) k_encoder(
    const float* __restrict__ x, const float* __restrict__ inter,
    const float* __restrict__ W1ft, const float* __restrict__ W2ft,
    const float* __restrict__ bf, float* __restrict__ hout) {
  int wave = threadIdx.x >> 5, lane = threadIdx.x & 31;
  int tile = blockIdx.x * 4 + wave;  // 2500 tiles
  int m = lane & 15, hi = lane >> 4;
  int arow = tile * 16 + m;
  for (int nt = 0; nt < 8; ++nt) {
    int col = nt * 16 + m;
    v8f acc = v8zero();
    for (int k0 = 0; k0 < FIN; k0 += 4) {
      int ka = k0 + 2 * hi;
      v2f a = {x[arow * FIN + ka], x[arow * FIN + ka + 1]};
      v2f b = {W1ft[col * FIN + ka], W1ft[col * FIN + ka + 1]};
      acc = wmma4(a, b, acc);
    }
    for (int k0 = 0; k0 < HH; k0 += 4) {
      int ka = k0 + 2 * hi;
      v2f a = {inter[arow * HH + ka], inter[arow * HH + ka + 1]};
      v2f b = {W2ft[col * HH + ka], W2ft[col * HH + ka + 1]};
      acc = wmma4(a, b, acc);
    }
#pragma unroll
    for (int r = 0; r < 8; ++r) {
      int orow = tile * 16 + r + 8 * hi;
      hout[orow * HH + col] = fmaxf(acc[r] + bf[col], 0.f);
    }
  }
}

// ----------------- fused bond-encoder + message + scatter -------------------
// per 32-edge wave tile (two 16-row A subtiles sharing every B fragment):
//   e = relu(EA@bw1+bb1)@bw2+bb2   (WMMA, LDS transpose)
//   m = relu(h[src]+e)*w ; atomicAdd -> aggr[dst]
__global__ void __launch_bounds__(64) k_bond_msg(
    const float* __restrict__ ea, const float* __restrict__ ewt,
    const int* __restrict__ src, const int* __restrict__ dst,
    const float* __restrict__ bw1t, const float* __restrict__ bb1,
    const float* __restrict__ bw2t, const float* __restrict__ bb2,
    const float* __restrict__ hcur, float* __restrict__ aggr) {
  __shared__ float Tls[2][32 * 132];
  int wave = threadIdx.x >> 5, lane = threadIdx.x & 31;
  int tile = blockIdx.x * 2 + wave;  // 20000 tiles of 32 edges
  int e0 = tile * 32;
  int m = lane & 15, hi = lane >> 4;
  float* T = &Tls[wave][0];

  // stage 1: (2x) 16x4 @ 4x128, single K-step per n-tile
#pragma unroll
  for (int s = 0; s < 2; ++s) {
    int er = e0 + 16 * s + m;
    for (int nt = 0; nt < 8; ++nt) {
      int col = nt * 16 + m;
      v2f a = {ea[er * 4 + 2 * hi], ea[er * 4 + 2 * hi + 1]};
      v2f b = {bw1t[col * 4 + 2 * hi], bw1t[col * 4 + 2 * hi + 1]};
      v8f acc = wmma4(a, b, v8zero());
#pragma unroll
      for (int r = 0; r < 8; ++r) {
        int rr = 16 * s + r + 8 * hi;
        T[rr * 132 + col] = fmaxf(acc[r] + bb1[col], 0.f);
      }
    }
  }
  __syncthreads();

  // stage 2: 32x128 @ 128x128 from LDS; B fragment reused by both subtiles
  for (int nt = 0; nt < 8; ++nt) {
    int col = nt * 16 + m;
    v8f acc0 = v8zero();
    v8f acc1 = v8zero();
    for (int k0 = 0; k0 < HH; k0 += 4) {
      int ka = k0 + 2 * hi;
      v2f b = {bw2t[col * HH + ka], bw2t[col * HH + ka + 1]};
      v2f a0 = {T[m * 132 + ka], T[m * 132 + ka + 1]};
      v2f a1 = {T[(16 + m) * 132 + ka], T[(16 + m) * 132 + ka + 1]};
      acc0 = wmma4(a0, b, acc0);
      acc1 = wmma4(a1, b, acc1);
    }
#pragma unroll
    for (int s = 0; s < 2; ++s) {
      v8f acc = s ? acc1 : acc0;
#pragma unroll
      for (int r = 0; r < 8; ++r) {
        int e = e0 + 16 * s + r + 8 * hi;
        int sn = src[e];
        int dn = dst[e];
        float w = ewt[e];
        float ev = acc[r] + bb2[col];
        float msg = fmaxf(hcur[sn * HH + col] + ev, 0.f) * w;
        atomicAdd(&aggr[dn * HH + col], msg);
      }
    }
  }
}

// ------------------- node update MLP (+ BN statistics) ----------------------
// z = relu(((1+eps)h + aggr)@w1 + b1) ; v = z@w2 + b2 ; accumulate sum/sum^2
// 32 nodes per wave; B fragments shared by both 16-row subtiles.
__global__ void __launch_bounds__(64) k_update(
    const float* __restrict__ hcur, const float* __restrict__ aggr,
    const float* __restrict__ epsp, const float* __restrict__ w1t,
    const float* __restrict__ b1, const float* __restrict__ w2t,
    const float* __restrict__ b2, float* __restrict__ vbuf,
    float* __restrict__ ssum, float* __restrict__ ssq) {
  __shared__ float Tls[2][32 * 132];
  int wave = threadIdx.x >> 5, lane = threadIdx.x & 31;
  int tile = blockIdx.x * 2 + wave;  // 1250 tiles of 32 nodes
  int n0 = tile * 32;
  int m = lane & 15, hi = lane >> 4;
  float* T = &Tls[wave][0];
  float epv = 1.f + epsp[0];

  // stage 1
  for (int nt = 0; nt < 8; ++nt) {
    int col = nt * 16 + m;
    v8f acc0 = v8zero();
    v8f acc1 = v8zero();
    for (int k0 = 0; k0 < HH; k0 += 4) {
      int ka = k0 + 2 * hi;
      v2f b = {w1t[col * HH + ka], w1t[col * HH + ka + 1]};
      int b0 = (n0 + m) * HH + ka;
      int b1i = (n0 + 16 + m) * HH + ka;
      v2f a0 = {epv * hcur[b0] + aggr[b0], epv * hcur[b0 + 1] + aggr[b0 + 1]};
      v2f a1 = {epv * hcur[b1i] + aggr[b1i],
                epv * hcur[b1i + 1] + aggr[b1i + 1]};
      acc0 = wmma4(a0, b, acc0);
      acc1 = wmma4(a1, b, acc1);
    }
#pragma unroll
    for (int s = 0; s < 2; ++s) {
      v8f acc = s ? acc1 : acc0;
#pragma unroll
      for (int r = 0; r < 8; ++r) {
        int rr = 16 * s + r + 8 * hi;
        T[rr * 132 + col] = fmaxf(acc[r] + b1[col], 0.f);
      }
    }
  }
  __syncthreads();

  // stage 2
  for (int nt = 0; nt < 8; ++nt) {
    int col = nt * 16 + m;
    v8f acc0 = v8zero();
    v8f acc1 = v8zero();
    for (int k0 = 0; k0 < HH; k0 += 4) {
      int ka = k0 + 2 * hi;
      v2f b = {w2t[col * HH + ka], w2t[col * HH + ka + 1]};
      v2f a0 = {T[m * 132 + ka], T[m * 132 + ka + 1]};
      v2f a1 = {T[(16 + m) * 132 + ka], T[(16 + m) * 132 + ka + 1]};
      acc0 = wmma4(a0, b, acc0);
      acc1 = wmma4(a1, b, acc1);
    }
    float sacc = 0.f, qacc = 0.f;
#pragma unroll
    for (int s = 0; s < 2; ++s) {
      v8f acc = s ? acc1 : acc0;
#pragma unroll
      for (int r = 0; r < 8; ++r) {
        int orow = n0 + 16 * s + r + 8 * hi;
        float v = acc[r] + b2[col];
        vbuf[orow * HH + col] = v;
        sacc += v;
        qacc += v * v;
      }
    }
    atomicAdd(&ssum[col], sacc);
    atomicAdd(&ssq[col], qacc);
  }
}

__global__ void k_bn_finalize(const float* __restrict__ ssum,
                              const float* __restrict__ ssq,
                              const float* __restrict__ gamma,
                              const float* __restrict__ beta,
                              float* __restrict__ scale,
                              float* __restrict__ shift) {
  int c = threadIdx.x;
  if (c < HH) {
    float mu = ssum[c] / (float)NN;
    float var = ssq[c] / (float)NN - mu * mu;
    float inv = rsqrtf(var + 1e-5f);
    float sc = gamma[c] * inv;
    scale[c] = sc;
    shift[c] = beta[c] - mu * sc;
  }
}

__global__ void k_bn_apply(const float* __restrict__ vbuf,
                           const float* __restrict__ scale,
                           const float* __restrict__ shift,
                           const float* __restrict__ hcur,
                           float* __restrict__ hnext, int use_res) {
  int i = blockIdx.x * blockDim.x + threadIdx.x;
  if (i < NN * HH) {
    int c = i & (HH - 1);
    float v = fmaxf(vbuf[i] * scale[c] + shift[c], 0.f);
    hnext[i] = use_res ? (hcur[i] + v) : v;
  }
}

// ------------------------------ pooling -------------------------------------

__global__ void k_pool(const float* __restrict__ h, const int* __restrict__ bat,
                       float* __restrict__ hg, float* __restrict__ cnt) {
  int i = blockIdx.x * blockDim.x + threadIdx.x;
  if (i < NN * HH) {
    int n = i >> 7, c = i & (HH - 1);
    int g = bat[n];
    atomicAdd(&hg[g * HH + c], h[i]);
    if (c == 0) atomicAdd(&cnt[g], 1.f);
  }
}

__global__ void k_pool_div(float* __restrict__ hg,
                           const float* __restrict__ cnt) {
  int i = blockIdx.x * blockDim.x + threadIdx.x;
  if (i < GG * HH) hg[i] /= fmaxf(cnt[i >> 7], 1.f);
}

// ------------------------------ head MLP ------------------------------------
// out = (relu(relu(hg@w0+c0)@w1+c1)) @ w2 + c2   (w0t/w1t pre-transposed)
__global__ void __launch_bounds__(128) k_head(
    const float* __restrict__ hg, const float* __restrict__ w0t,
    const float* __restrict__ c0, const float* __restrict__ w1t,
    const float* __restrict__ c1, const float* __restrict__ w2,
    const float* __restrict__ c2, float* __restrict__ out) {
  __shared__ float Tls[4][16 * 132];
  int wave = threadIdx.x >> 5, lane = threadIdx.x & 31;
  int tile = blockIdx.x * 4 + wave;  // 64 tiles
  int g0 = tile * 16;
  int m = lane & 15, hi = lane >> 4;
  float* T = &Tls[wave][0];

  // layer 0
  for (int nt = 0; nt < 8; ++nt) {
    int col = nt * 16 + m;
    v8f acc = v8zero();
    for (int k0 = 0; k0 < HH; k0 += 4) {
      int ka = k0 + 2 * hi;
      v2f a = {hg[(g0 + m) * HH + ka], hg[(g0 + m) * HH + ka + 1]};
      v2f b = {w0t[col * HH + ka], w0t[col * HH + ka + 1]};
      acc = wmma4(a, b, acc);
    }
#pragma unroll
    for (int r = 0; r < 8; ++r)
      T[(r + 8 * hi) * 132 + col] = fmaxf(acc[r] + c0[col], 0.f);
  }
  __syncthreads();

  // layer 1: read whole tile first, then overwrite in place
  v8f accs[8];
  for (int nt = 0; nt < 8; ++nt) {
    v8f acc = v8zero();
    int col = nt * 16 + m;
    for (int k0 = 0; k0 < HH; k0 += 4) {
      int ka = k0 + 2 * hi;
      v2f a = {T[m * 132 + ka], T[m * 132 + ka + 1]};
      v2f b = {w1t[col * HH + ka], w1t[col * HH + ka + 1]};
      acc = wmma4(a, b, acc);
    }
    accs[nt] = acc;
  }
  __syncthreads();
  for (int nt = 0; nt < 8; ++nt) {
    int col = nt * 16 + m;
#pragma unroll
    for (int r = 0; r < 8; ++r)
      T[(r + 8 * hi) * 132 + col] = fmaxf(accs[nt][r] + c1[col], 0.f);
  }
  __syncthreads();

  // layer 2: H -> 1 dot product, one graph per lane (lanes 0..15)
  if (lane < 16) {
    float s = c2[0];
    for (int c = 0; c < HH; ++c) s += T[lane * 132 + c] * w2[c];
    out[g0 + lane] = s;
  }
}

// ---------------------------------------------------------------------------

extern "C" void kernel_launch(void* const* d_in, const int* in_sizes, int n_in,
                              void* d_out, int out_size, void* d_ws,
                              size_t ws_size, hipStream_t stream) {
  (void)in_sizes; (void)n_in; (void)out_size; (void)ws_size;

  const float* x     = (const float*)d_in[0];
  const float* inter = (const float*)d_in[1];
  const float* ea    = (const float*)d_in[2];
  const float* ewt   = (const float*)d_in[3];
  const int*   eidx  = (const int*)d_in[4];
  const int*   bat   = (const int*)d_in[5];
  const float* aw = (const float*)d_in[6];
  const float* ab = (const float*)d_in[7];
  const float* ew = (const float*)d_in[8];
  const float* eb = (const float*)d_in[9];
  const float* mw = (const float*)d_in[10];
  const float* mb = (const float*)d_in[11];
  // convs: 4 x {eps,w1,b1,w2,b2,gamma,beta,bw1,bb1,bw2,bb2} starting at 12
  const int C0 = 12, CS = 11;
  const float* hw0 = (const float*)d_in[C0 + 4 * CS + 0];
  const float* hw1 = (const float*)d_in[C0 + 4 * CS + 1];
  const float* hw2 = (const float*)d_in[C0 + 4 * CS + 2];
  const float* hb0 = (const float*)d_in[C0 + 4 * CS + 3];
  const float* hb1 = (const float*)d_in[C0 + 4 * CS + 4];
  const float* hb2 = (const float*)d_in[C0 + 4 * CS + 5];

  const int* srcp = eidx;
  const int* dstp = eidx + EE;

  // workspace layout (floats)
  float* ws = (float*)d_ws;
  float* W1ft = ws;                 ws += FIN * HH;
  float* W2ft = ws;                 ws += HH * HH;
  float* bf   = ws;                 ws += HH;
  float* hA   = ws;                 ws += NN * HH;
  float* hB   = ws;                 ws += NN * HH;
  float* aggr = ws;                 ws += NN * HH;
  float* vbuf = ws;                 ws += NN * HH;
  float* ssum = ws;                 ws += HH;
  float* ssq  = ws;                 ws += HH;
  float* scl  = ws;                 ws += HH;
  float* shf  = ws;                 ws += HH;
  float* hg   = ws;                 ws += GG * HH;
  float* cnt  = ws;                 ws += GG;
  float* bw1t = ws;                 ws += 4 * HH;
  float* bw2t = ws;                 ws += HH * HH;
  float* w1t  = ws;                 ws += HH * HH;
  float* w2t  = ws;                 ws += HH * HH;
  float* hw0t = ws;                 ws += HH * HH;
  float* hw1t = ws;                 ws += HH * HH;

  // encoder
  k_fuse_weights<<<(FIN * HH + HH * HH + HH + 255) / 256, 256, 0, stream>>>(
      aw, ab, ew, eb, mw, mb, W1ft, W2ft, bf);
  k_encoder<<<NN / 64, 128, 0, stream>>>(x, inter, W1ft, W2ft, bf, hA);

  float* hcur = hA;
  float* hnext = hB;
  for (int l = 0; l < 4; ++l) {
    const float* eps = (const float*)d_in[C0 + l * CS + 0];
    const float* w1  = (const float*)d_in[C0 + l * CS + 1];
    const float* b1  = (const float*)d_in[C0 + l * CS + 2];
    const float* w2  = (const float*)d_in[C0 + l * CS + 3];
    const float* b2  = (const float*)d_in[C0 + l * CS + 4];
    const float* gam = (const float*)d_in[C0 + l * CS + 5];
    const float* bet = (const float*)d_in[C0 + l * CS + 6];
    const float* bw1 = (const float*)d_in[C0 + l * CS + 7];
    const float* bb1 = (const float*)d_in[C0 + l * CS + 8];
    const float* bw2 = (const float*)d_in[C0 + l * CS + 9];
    const float* bb2 = (const float*)d_in[C0 + l * CS + 10];

    // per-layer weight transposes (tiny)
    k_transpose<<<(4 * HH + 255) / 256, 256, 0, stream>>>(bw1, bw1t, 4, HH);
    k_transpose<<<(HH * HH + 255) / 256, 256, 0, stream>>>(bw2, bw2t, HH, HH);
    k_transpose<<<(HH * HH + 255) / 256, 256, 0, stream>>>(w1, w1t, HH, HH);
    k_transpose<<<(HH * HH + 255) / 256, 256, 0, stream>>>(w2, w2t, HH, HH);

    k_zero<<<(NN * HH + 255) / 256, 256, 0, stream>>>(aggr, NN * HH);
    k_zero<<<1, 256, 0, stream>>>(ssum, 2 * HH);  // ssum+ssq contiguous
    // 20000 tiles of 32 edges, 2 waves (tiles) per 64-thread block
    k_bond_msg<<<EE / 64, 64, 0, stream>>>(ea, ewt, srcp, dstp, bw1t, bb1, bw2t,
                                           bb2, hcur, aggr);
    // 1250 tiles of 32 nodes, 2 waves per block
    k_update<<<NN / 64, 64, 0, stream>>>(hcur, aggr, eps, w1t, b1, w2t, b2,
                                         vbuf, ssum, ssq);
    k_bn_finalize<<<1, 128, 0, stream>>>(ssum, ssq, gam, bet, scl, shf);
    k_bn_apply<<<(NN * HH + 255) / 256, 256, 0, stream>>>(vbuf, scl, shf, hcur,
                                                          hnext, l > 0 ? 1 : 0);
    float* t = hcur; hcur = hnext; hnext = t;
  }

  // pooling + head
  k_zero<<<(GG * HH + GG + 255) / 256, 256, 0, stream>>>(hg, GG * HH + GG);
  k_pool<<<(NN * HH + 255) / 256, 256, 0, stream>>>(hcur, bat, hg, cnt);
  k_pool_div<<<(GG * HH + 255) / 256, 256, 0, stream>>>(hg, cnt);
  k_transpose<<<(HH * HH + 255) / 256, 256, 0, stream>>>(hw0, hw0t, HH, HH);
  k_transpose<<<(HH * HH + 255) / 256, 256, 0, stream>>>(hw1, hw1t, HH, HH);
  k_head<<<GG / 64, 128, 0, stream>>>(hg, hw0t, hb0, hw1t, hb1, hw2, hb2,
                                      (float*)d_out);
}